// SimpleModelWithCache_50130858279076
// MI455X (gfx1250) — compile-verified
//
#include <hip/hip_runtime.h>
#include <hip/hip_bf16.h>

// ---------------------------------------------------------------------------
// Model dims (fixed by reference)
// ---------------------------------------------------------------------------
#define VV   32000
#define DD   1024
#define HH   16
#define LL   4
#define BB   8
#define QQ   8
#define PP   2048
#define HDIM 64
#define KTOT (PP + QQ)          // 2056
#define MROWS (BB * QQ)         // 64 token rows
#define CHUNK 128               // keys per streaming chunk (2048 = 16 * 128)
#define BPAD 68                 // LDS row stride (floats): 16B aligned, low-conflict

typedef __attribute__((ext_vector_type(2))) float v2f;
typedef __attribute__((ext_vector_type(8))) float v8f;

// ---------------------------------------------------------------------------
// Embedding gather: x[row] = embed[ids[row]]   (64 x 1024 f32)
// ---------------------------------------------------------------------------
__global__ __launch_bounds__(256) void embed_kernel(const int* __restrict__ ids,
                                                    const float* __restrict__ embed,
                                                    float* __restrict__ x) {
    const int row = blockIdx.x;                 // 0..63
    const int c4  = threadIdx.x * 4;            // 256 threads * 4 = 1024
    const int tok = ids[row];
    *(float4*)(x + (size_t)row * DD + c4) =
        *(const float4*)(embed + (size_t)tok * DD + c4);
}

// ---------------------------------------------------------------------------
// GEMM: Y[64,N] = X[64,1024] @ W[1024,N] + bias[N]   (full fp32 via WMMA)
// N is a compile-time template parameter so B-fragment strides become
// 24-bit immediate instruction offsets (kk*N*4 <= 7.68MB < 2^23 even for
// N=32000), eliminating per-load 64-bit address arithmetic.
// grid = (N/256, 64/16), block = 256 (8 wave32s), 2 tiles per wave.
// ---------------------------------------------------------------------------
template <int N>
__global__ __launch_bounds__(256) void gemm_kernel(const float* __restrict__ X,
                                                   const float* __restrict__ W,
                                                   const float* __restrict__ bias,
                                                   float* __restrict__ Y) {
    __shared__ float xs[16][66];                // padded: avoids bank conflicts

    const int t     = threadIdx.x;
    const int lane  = t & 31;
    const int wave  = t >> 5;
    const int row   = lane & 15;                // A row / B-N col within tile
    const int koff  = (lane >> 4) << 1;         // 0 or 2
    const int mbase = blockIdx.y * 16;
    const int nb0   = blockIdx.x * 256 + wave * 32;  // tile0 at nb0, tile1 at nb0+16
    const int K     = DD;

    v8f c0 = {0.f,0.f,0.f,0.f,0.f,0.f,0.f,0.f};
    v8f c1 = c0;

    // per-lane W base pointer; advances once per 64-k block
    const float* wbase = W + (size_t)koff * N + nb0 + row;

    for (int kb = 0; kb < K; kb += 64) {
        // cooperative stage of X[mbase..+16)[kb..+64) into LDS
        {
            const int r  = t >> 4;
            const int c4 = (t & 15) * 4;
            float4 v = *(const float4*)(X + (size_t)(mbase + r) * K + kb + c4);
            xs[r][c4 + 0] = v.x; xs[r][c4 + 1] = v.y;
            xs[r][c4 + 2] = v.z; xs[r][c4 + 3] = v.w;
        }
        __syncthreads();

        #pragma unroll
        for (int kk = 0; kk < 64; kk += 4) {
            v2f a;
            a.x = xs[row][kk + koff];
            a.y = xs[row][kk + koff + 1];
            v2f b0, b1;
            b0.x = wbase[(size_t)kk * N];            b0.y = wbase[(size_t)(kk + 1) * N];
            b1.x = wbase[(size_t)kk * N + 16];       b1.y = wbase[(size_t)(kk + 1) * N + 16];
            c0 = __builtin_amdgcn_wmma_f32_16x16x4_f32(false, a, false, b0,
                                                       (short)0, c0, false, false);
            c1 = __builtin_amdgcn_wmma_f32_16x16x4_f32(false, a, false, b1,
                                                       (short)0, c1, false, false);
        }
        __syncthreads();
        wbase += (size_t)64 * N;
    }

    // store D + bias.  D layout: vgpr r -> M = r + 8*(lane>=16), N = lane&15
    const int mo = (lane >> 4) * 8;
    const int n0 = nb0 + row;
    const int n1 = nb0 + 16 + row;
    const float bi0 = bias[n0];
    const float bi1 = bias[n1];
    #pragma unroll
    for (int r = 0; r < 8; ++r) {
        Y[(size_t)(mbase + mo + r) * N + n0] = c0[r] + bi0;
        Y[(size_t)(mbase + mo + r) * N + n1] = c1[r] + bi1;
    }
}

// ---------------------------------------------------------------------------
// Fused attention + KV-cache materialization for one layer.
// grid = B*H = 128 blocks, block = 256 threads (8 wave32s).
//  - past K/V chunks staged global->LDS with global_load_async_to_lds_b128
//    and copied LDS->output cache with global_store_async_from_lds_b128
//    (zero VGPR traffic for the 537 MB cache move; ASYNCcnt-tracked).
//  - Q.K^T scores computed with v_wmma_f32_16x16x4_f32 (q padded 8->16 rows).
//  - online softmax: wave w owns query row w; P.V accumulation on VALU with
//    persistent per-(q,d) accumulators.
// ---------------------------------------------------------------------------
__global__ __launch_bounds__(256) void attn_kernel(const float* __restrict__ qb,   // [64,1024]
                                                   const float* __restrict__ knew, // [64,1024]
                                                   const float* __restrict__ vnew, // [64,1024]
                                                   const float* __restrict__ pk,   // [B,H,P,HD]
                                                   const float* __restrict__ pv,
                                                   float* __restrict__ ok,         // [B,H,P+Q,HD]
                                                   float* __restrict__ ov,
                                                   float* __restrict__ ao) {       // [64,1024]
    __shared__ __align__(16) float qs[16][BPAD];     // padded q tile (rows 8..15 = 0)
    __shared__ __align__(16) float buf[CHUNK][BPAD]; // K or V chunk
    __shared__ float sb[QQ][CHUNK];                  // scores -> probabilities
    __shared__ float mrow[QQ], lrow[QQ], csc[QQ];

    const int t    = threadIdx.x;
    const int bh   = blockIdx.x;             // 0..127
    const int b    = bh >> 4;
    const int h    = bh & 15;
    const int lane = t & 31;
    const int wave = t >> 5;                 // softmax: wave == query row
    const int col  = lane & 15;
    const int koff = (lane >> 4) << 1;       // 0 or 2
    const int d    = t & 63;                 // accumulator dim
    const int q0   = t >> 6;                 // accumulator rows q0, q0+4
    const int q1   = q0 + 4;

    // load queries pre-scaled by 1/sqrt(64); zero-pad rows 8..15 for WMMA
    {
        const int r  = t >> 4;               // 0..15
        const int c4 = (t & 15) * 4;
        if (t < 128) {
            float4 v = *(const float4*)(qb + (size_t)(b * QQ + r) * DD + h * HDIM + c4);
            qs[r][c4 + 0] = v.x * 0.125f; qs[r][c4 + 1] = v.y * 0.125f;
            qs[r][c4 + 2] = v.z * 0.125f; qs[r][c4 + 3] = v.w * 0.125f;
        } else {
            qs[r][c4 + 0] = 0.f; qs[r][c4 + 1] = 0.f;
            qs[r][c4 + 2] = 0.f; qs[r][c4 + 3] = 0.f;
        }
    }
    if (t < QQ) { mrow[t] = -__builtin_inff(); lrow[t] = 0.f; }
    float acc0 = 0.f, acc1 = 0.f;
    __syncthreads();

    const float* pkbh = pk + (size_t)bh * PP * HDIM;
    const float* pvbh = pv + (size_t)bh * PP * HDIM;
    float* okbh = ok + (size_t)bh * KTOT * HDIM;
    float* ovbh = ov + (size_t)bh * KTOT * HDIM;

    const int NPC = PP / CHUNK;              // 16 past chunks (+1 for fresh keys)
    for (int c = 0; c <= NPC; ++c) {
        const int base = c * CHUNK;
        const int len  = (c < NPC) ? CHUNK : QQ;

        // drain this thread's outstanding async stores before reusing its
        // LDS region (each thread's async ops cover exactly its own region)
        asm volatile("s_wait_asynccnt 0x0" ::: "memory");

        // ---- stage K chunk into LDS ----
        if (c < NPC) {
            #pragma unroll
            for (int i = 0; i < (CHUNK * HDIM) / (256 * 4); ++i) {
                const int idx = (i * 256 + t) * 4;
                const int r = idx >> 6, cc = idx & 63;
                unsigned int       la = (unsigned int)(unsigned long long)&buf[r][cc];
                unsigned long long ga = (unsigned long long)(pkbh + (size_t)base * HDIM + idx);
                asm volatile("global_load_async_to_lds_b128 %0, %1, off"
                             :: "v"(la), "v"(ga) : "memory");
            }
            asm volatile("s_wait_asynccnt 0x0" ::: "memory");
        } else if (t < (QQ * HDIM) / 4) {
            const int idx = t * 4;
            const int r = idx >> 6, cc = idx & 63;
            float4 v = *(const float4*)(knew + (size_t)(b * QQ + r) * DD + h * HDIM + cc);
            *(float4*)&buf[r][cc] = v;
            *(float4*)(okbh + (size_t)(PP + r) * HDIM + cc) = v;
        }
        __syncthreads();

        // ---- fused copy-out of past K chunk: LDS -> output cache ----
        if (c < NPC) {
            #pragma unroll
            for (int i = 0; i < (CHUNK * HDIM) / (256 * 4); ++i) {
                const int idx = (i * 256 + t) * 4;
                const int r = idx >> 6, cc = idx & 63;
                unsigned int       la = (unsigned int)(unsigned long long)&buf[r][cc];
                unsigned long long ga = (unsigned long long)(okbh + (size_t)base * HDIM + idx);
                asm volatile("global_store_async_from_lds_b128 %0, %1, off"
                             :: "v"(ga), "v"(la) : "memory");
            }
        }

        // ---- scores via WMMA: S[16 x 16] tile per wave, K-dim = 64 ----
        {
            v8f sc = {0.f,0.f,0.f,0.f,0.f,0.f,0.f,0.f};
            const int jbase = wave * 16;
            #pragma unroll
            for (int dk = 0; dk < HDIM; dk += 4) {
                v2f a, bf;
                a.x  = qs[col][dk + koff];
                a.y  = qs[col][dk + koff + 1];
                bf.x = buf[jbase + col][dk + koff];
                bf.y = buf[jbase + col][dk + koff + 1];
                sc = __builtin_amdgcn_wmma_f32_16x16x4_f32(false, a, false, bf,
                                                           (short)0, sc, false, false);
            }
            // valid rows (q=0..7) live in lanes 0..15, vgprs 0..7
            if (lane < 16) {
                #pragma unroll
                for (int r = 0; r < 8; ++r) sb[r][jbase + lane] = sc[r];
            }
        }
        __syncthreads();

        // ---- online softmax: wave w reduces row w from sb ----
        {
            float s[4];
            #pragma unroll
            for (int u = 0; u < 4; ++u) {
                const int j = lane + u * 32;
                s[u] = (j < len) ? sb[wave][j] : -__builtin_inff();
            }
            float cmax = fmaxf(fmaxf(s[0], s[1]), fmaxf(s[2], s[3]));
            #pragma unroll
            for (int off = 16; off; off >>= 1) cmax = fmaxf(cmax, __shfl_xor(cmax, off, 32));
            const float oldm = mrow[wave];
            const float newm = fmaxf(oldm, cmax);
            float psum = 0.f;
            #pragma unroll
            for (int u = 0; u < 4; ++u) {
                const int j = lane + u * 32;
                const float p = (j < len) ? __expf(s[u] - newm) : 0.f;
                sb[wave][j] = p;
                psum += p;
            }
            #pragma unroll
            for (int off = 16; off; off >>= 1) psum += __shfl_xor(psum, off, 32);
            if (lane == 0) {
                const float scl = __expf(oldm - newm);
                csc[wave]  = scl;
                lrow[wave] = lrow[wave] * scl + psum;
                mrow[wave] = newm;
            }
        }
        __syncthreads();                       // probs + csc ready; K reads done

        acc0 *= csc[q0];
        acc1 *= csc[q1];

        // drain K copy-out stores before overwriting buf with V
        asm volatile("s_wait_asynccnt 0x0" ::: "memory");

        // ---- stage V chunk into LDS ----
        if (c < NPC) {
            #pragma unroll
            for (int i = 0; i < (CHUNK * HDIM) / (256 * 4); ++i) {
                const int idx = (i * 256 + t) * 4;
                const int r = idx >> 6, cc = idx & 63;
                unsigned int       la = (unsigned int)(unsigned long long)&buf[r][cc];
                unsigned long long ga = (unsigned long long)(pvbh + (size_t)base * HDIM + idx);
                asm volatile("global_load_async_to_lds_b128 %0, %1, off"
                             :: "v"(la), "v"(ga) : "memory");
            }
            asm volatile("s_wait_asynccnt 0x0" ::: "memory");
        } else if (t < (QQ * HDIM) / 4) {
            const int idx = t * 4;
            const int r = idx >> 6, cc = idx & 63;
            float4 v = *(const float4*)(vnew + (size_t)(b * QQ + r) * DD + h * HDIM + cc);
            *(float4*)&buf[r][cc] = v;
            *(float4*)(ovbh + (size_t)(PP + r) * HDIM + cc) = v;
        }
        __syncthreads();

        // ---- fused copy-out of past V chunk: LDS -> output cache ----
        if (c < NPC) {
            #pragma unroll
            for (int i = 0; i < (CHUNK * HDIM) / (256 * 4); ++i) {
                const int idx = (i * 256 + t) * 4;
                const int r = idx >> 6, cc = idx & 63;
                unsigned int       la = (unsigned int)(unsigned long long)&buf[r][cc];
                unsigned long long ga = (unsigned long long)(ovbh + (size_t)base * HDIM + idx);
                asm volatile("global_store_async_from_lds_b128 %0, %1, off"
                             :: "v"(ga), "v"(la) : "memory");
            }
        }

        // ---- accumulate: acc[q][d] += p[q][j] * v[j][d] ----
        for (int j = 0; j < len; ++j) {
            const float vv = buf[j][d];
            acc0 += sb[q0][j] * vv;
            acc1 += sb[q1][j] * vv;
        }
        __syncthreads();                       // before next chunk overwrites buf
    }

    ao[(size_t)(b * QQ + q0) * DD + h * HDIM + d] = acc0 / lrow[q0];
    ao[(size_t)(b * QQ + q1) * DD + h * HDIM + d] = acc1 / lrow[q1];
    // s_endpgm performs an implicit wait-idle for outstanding async stores
}

// ---------------------------------------------------------------------------
// Host launcher
// ---------------------------------------------------------------------------
extern "C" void kernel_launch(void* const* d_in, const int* in_sizes, int n_in,
                              void* d_out, int out_size, void* d_ws, size_t ws_size,
                              hipStream_t stream) {
    const int*   ids    = (const int*)  d_in[0];
    const float* past_k = (const float*)d_in[1];
    const float* past_v = (const float*)d_in[2];
    const float* embed  = (const float*)d_in[3];
    const float* q_w    = (const float*)d_in[4];
    const float* q_b    = (const float*)d_in[5];
    const float* k_w    = (const float*)d_in[6];
    const float* k_b    = (const float*)d_in[7];
    const float* v_w    = (const float*)d_in[8];
    const float* v_b    = (const float*)d_in[9];
    const float* o_w    = (const float*)d_in[10];
    const float* o_b    = (const float*)d_in[11];
    const float* lm_w   = (const float*)d_in[12];
    const float* lm_b   = (const float*)d_in[13];

    float* out    = (float*)d_out;
    float* logits = out;                                    // [64, V]
    float* new_ks = out + (size_t)MROWS * VV;               // [L,B,H,KTOT,HD]
    float* new_vs = new_ks + (size_t)LL * BB * HH * KTOT * HDIM;

    // workspace layout (fp32)
    float* ws   = (float*)d_ws;
    float* x0   = ws;                        // [64,1024]
    float* x1   = x0 + MROWS * DD;
    float* qbuf = x1 + MROWS * DD;
    float* kbuf = qbuf + MROWS * DD;
    float* vbuf = kbuf + MROWS * DD;
    float* aob  = vbuf + MROWS * DD;

    // 1) embedding gather
    embed_kernel<<<dim3(MROWS), dim3(256), 0, stream>>>(ids, embed, x0);

    float* x_cur = x0;
    float* x_nxt = x1;
    const dim3 gproj(DD / 256, MROWS / 16);                 // (4, 4)
    const size_t wstride = (size_t)DD * DD;                 // per-layer weight stride
    const size_t pstride = (size_t)BB * HH * PP * HDIM;     // past KV per-layer stride
    const size_t ostride = (size_t)BB * HH * KTOT * HDIM;   // out KV per-layer stride

    for (int l = 0; l < LL; ++l) {
        gemm_kernel<DD><<<gproj, dim3(256), 0, stream>>>(x_cur, q_w + l * wstride,
                                                         q_b + l * DD, qbuf);
        gemm_kernel<DD><<<gproj, dim3(256), 0, stream>>>(x_cur, k_w + l * wstride,
                                                         k_b + l * DD, kbuf);
        gemm_kernel<DD><<<gproj, dim3(256), 0, stream>>>(x_cur, v_w + l * wstride,
                                                         v_b + l * DD, vbuf);

        attn_kernel<<<dim3(BB * HH), dim3(256), 0, stream>>>(
            qbuf, kbuf, vbuf,
            past_k + l * pstride, past_v + l * pstride,
            new_ks + l * ostride, new_vs + l * ostride, aob);

        gemm_kernel<DD><<<gproj, dim3(256), 0, stream>>>(aob, o_w + l * wstride,
                                                         o_b + l * DD, x_nxt);
        float* tmp = x_cur; x_cur = x_nxt; x_nxt = tmp;
    }

    // 2) LM head: [64,1024] @ [1024,32000] + bias
    gemm_kernel<VV><<<dim3(VV / 256, MROWS / 16), dim3(256), 0, stream>>>(
        x_cur, lm_w, lm_b, logits);
}